// LinearPerformerAttention_68427418959957
// MI455X (gfx1250) — compile-verified
//
#include <hip/hip_runtime.h>
#include <hip/hip_bf16.h>

// ---------------------------------------------------------------------------
// Causal linear (Performer) attention, fp32, via V_WMMA_F32_16X16X4_F32.
// Chunked-scan formulation; every matmul stage on fp32 WMMA. This revision
// widens each wave's tile (1 A-fragment feeds 4 (or 2) B-fragments/WMMAs) and
// makes all strides compile-time so loads use immediate offsets.
// ---------------------------------------------------------------------------

typedef __attribute__((ext_vector_type(2))) float v2f;
typedef __attribute__((ext_vector_type(8))) float v8f;

#define DIMX 512
#define HEADS 8
#define HDIM 64
#define FEAT 128
#define NTOK 2048
#define QKV_LD (3 * DIMX)
#define CHUNK 64
#define NCHUNK (NTOK / CHUNK)
#define EPSV 1e-8f

__device__ __forceinline__ v8f wmma4(v2f a, v2f b, v8f c) {
  // D = A(16x4 f32) * B(4x16 f32) + C(16x16 f32)
  return __builtin_amdgcn_wmma_f32_16x16x4_f32(
      false, a, false, b, (short)0, c, false, false);
}

__device__ __forceinline__ void store_tile(float* __restrict__ C, int ldc,
                                           int row0, int col, v8f acc,
                                           float bb) {
#pragma unroll
  for (int i = 0; i < 8; ++i) C[(size_t)(row0 + i) * ldc + col] = acc[i] + bb;
}

// Generic row-major fp32 GEMM: C[M,N] = A[M,K] @ B[K,N] (+ bias[N]).
// One 16x64 output tile per wave (4 accumulators share one A fragment).
template <int M, int N, int K, int LDA, int LDB, int LDC, bool BIAS>
__global__ __launch_bounds__(256) void gemm_wmma_f32(
    const float* __restrict__ A, const float* __restrict__ B,
    float* __restrict__ C, const float* __restrict__ bias) {
  constexpr int TN = N / 64;
  const int wave = (blockIdx.x * 256 + threadIdx.x) >> 5;
  const int lane = threadIdx.x & 31;
  const int tm = (wave / TN) << 4;
  const int tn0 = (wave % TN) << 6;
  if (tm >= M) return;
  const int half = lane >> 4;  // 0: K+0,K+1   1: K+2,K+3
  const int l = lane & 15;
  const float* ap = A + (size_t)(tm + l) * LDA + 2 * half;
  const float* bp = B + (size_t)(2 * half) * LDB + tn0 + l;
  v8f acc0 = {}, acc1 = {}, acc2 = {}, acc3 = {};
  for (int k = 0; k < K; k += 4) {
    v2f a, b0, b1, b2, b3;
    a.x = ap[0];
    a.y = ap[1];
    b0.x = bp[0];
    b0.y = bp[LDB];
    b1.x = bp[16];
    b1.y = bp[LDB + 16];
    b2.x = bp[32];
    b2.y = bp[LDB + 32];
    b3.x = bp[48];
    b3.y = bp[LDB + 48];
    acc0 = wmma4(a, b0, acc0);
    acc1 = wmma4(a, b1, acc1);
    acc2 = wmma4(a, b2, acc2);
    acc3 = wmma4(a, b3, acc3);
    ap += 4;
    bp += (size_t)4 * LDB;
  }
  const int row0 = tm + 8 * half;
  const int col = tn0 + l;
  store_tile(C, LDC, row0, col, acc0, BIAS ? bias[col] : 0.0f);
  store_tile(C, LDC, row0, col + 16, acc1, BIAS ? bias[col + 16] : 0.0f);
  store_tile(C, LDC, row0, col + 32, acc2, BIAS ? bias[col + 32] : 0.0f);
  store_tile(C, LDC, row0, col + 48, acc3, BIAS ? bias[col + 48] : 0.0f);
}

// Feature map: for b16 = (sel<<3)|h, compute elu(Q_or_K_head @ proj_h)+1.
// M=2048, N=128, K=64; one 16x64 tile per wave.  elu(x)+1 = x>0 ? x+1 : exp(x)
__global__ __launch_bounds__(256) void featmap_wmma(
    const float* __restrict__ qkv, const float* __restrict__ proj,
    float* __restrict__ qp, float* __restrict__ kp) {
  const int b16 = blockIdx.y;
  const int sel = b16 >> 3;  // 0 = Q, 1 = K
  const int h = b16 & 7;
  const float* A = qkv + sel * DIMX + h * HDIM;     // lda = QKV_LD
  const float* B = proj + (size_t)h * HDIM * FEAT;  // ldb = FEAT
  float* out = (sel ? kp : qp) + (size_t)h * NTOK * FEAT;
  const int wave = (blockIdx.x * 256 + threadIdx.x) >> 5;
  const int lane = threadIdx.x & 31;
  const int tm = (wave >> 1) << 4;
  const int tn0 = (wave & 1) << 6;
  const int half = lane >> 4, l = lane & 15;
  const float* ap = A + (size_t)(tm + l) * QKV_LD + 2 * half;
  const float* bp = B + (size_t)(2 * half) * FEAT + tn0 + l;
  v8f acc0 = {}, acc1 = {}, acc2 = {}, acc3 = {};
  for (int k = 0; k < HDIM; k += 4) {
    v2f a, b0, b1, b2, b3;
    a.x = ap[0];
    a.y = ap[1];
    b0.x = bp[0];
    b0.y = bp[FEAT];
    b1.x = bp[16];
    b1.y = bp[FEAT + 16];
    b2.x = bp[32];
    b2.y = bp[FEAT + 32];
    b3.x = bp[48];
    b3.y = bp[FEAT + 48];
    acc0 = wmma4(a, b0, acc0);
    acc1 = wmma4(a, b1, acc1);
    acc2 = wmma4(a, b2, acc2);
    acc3 = wmma4(a, b3, acc3);
    ap += 4;
    bp += 4 * FEAT;
  }
  const int row0 = tm + 8 * half;
#pragma unroll
  for (int j = 0; j < 4; ++j) {
    v8f acc = (j == 0) ? acc0 : (j == 1) ? acc1 : (j == 2) ? acc2 : acc3;
    const int col = tn0 + 16 * j + l;
#pragma unroll
    for (int i = 0; i < 8; ++i) {
      float t = acc[i];
      out[(size_t)(row0 + i) * FEAT + col] =
          (t > 0.0f) ? (t + 1.0f) : __expf(t);
    }
  }
}

// Per-chunk S_c[f][d] = sum_{n in chunk} kp[n][f]*v[n][d]  (128x64 = Kp^T @ V)
// One block per (chunk, head); 8 waves, each a 16(f) x 64(d) tile.
__global__ __launch_bounds__(256) void chunk_kv_wmma(
    const float* __restrict__ kp, const float* __restrict__ qkv,
    float* __restrict__ Ssum) {
  const int c = blockIdx.x, h = blockIdx.y;
  const int wave = threadIdx.x >> 5;  // 0..7 -> f tile
  const int lane = threadIdx.x & 31;
  const int tm = wave << 4;
  const int half = lane >> 4, l = lane & 15;
  const int c0 = c * CHUNK;
  const float* kpB = kp + ((size_t)h * NTOK + c0) * FEAT;             // [n][f]
  const float* vB = qkv + (size_t)c0 * QKV_LD + 2 * DIMX + h * HDIM;  // [n][d]
  const float* ap = kpB + (size_t)(2 * half) * FEAT + tm + l;  // A[f][n]=kp[n][f]
  const float* bp = vB + (size_t)(2 * half) * QKV_LD + l;
  v8f acc0 = {}, acc1 = {}, acc2 = {}, acc3 = {};
  for (int k = 0; k < CHUNK; k += 4) {
    v2f a, b0, b1, b2, b3;
    a.x = ap[0];
    a.y = ap[FEAT];
    b0.x = bp[0];
    b0.y = bp[QKV_LD];
    b1.x = bp[16];
    b1.y = bp[QKV_LD + 16];
    b2.x = bp[32];
    b2.y = bp[QKV_LD + 32];
    b3.x = bp[48];
    b3.y = bp[QKV_LD + 48];
    acc0 = wmma4(a, b0, acc0);
    acc1 = wmma4(a, b1, acc1);
    acc2 = wmma4(a, b2, acc2);
    acc3 = wmma4(a, b3, acc3);
    ap += 4 * FEAT;
    bp += 4 * QKV_LD;
  }
  float* out = Ssum + (size_t)(h * NCHUNK + c) * FEAT * HDIM;
  const int row0 = tm + 8 * half;
  store_tile(out, HDIM, row0, l, acc0, 0.0f);
  store_tile(out, HDIM, row0, l + 16, acc1, 0.0f);
  store_tile(out, HDIM, row0, l + 32, acc2, 0.0f);
  store_tile(out, HDIM, row0, l + 48, acc3, 0.0f);
}

// z_c[f] = sum_{n in chunk} kp[n][f]
__global__ void chunk_z(const float* __restrict__ kp,
                        float* __restrict__ zsum) {
  const int idx = blockIdx.x * blockDim.x + threadIdx.x;
  if (idx >= HEADS * NCHUNK * FEAT) return;
  const int f = idx % FEAT;
  const int c = (idx / FEAT) % NCHUNK;
  const int h = idx / (FEAT * NCHUNK);
  const float* base = kp + ((size_t)h * NTOK + c * CHUNK) * FEAT + f;
  float s = 0.0f;
  for (int n = 0; n < CHUNK; ++n) s += base[(size_t)n * FEAT];
  zsum[idx] = s;
}

// Exclusive prefix over chunks (in place); each thread owns one element.
__global__ void prefix_chunks(float* __restrict__ Ssum,
                              float* __restrict__ zsum) {
  const int h = blockIdx.y;
  const int e = blockIdx.x * blockDim.x + threadIdx.x;
  const int SZ = FEAT * HDIM;  // 8192
  if (e < SZ) {
    float run = 0.0f;
    for (int c = 0; c < NCHUNK; ++c) {
      const size_t ix = (size_t)(h * NCHUNK + c) * SZ + e;
      const float t = Ssum[ix];
      Ssum[ix] = run;
      run += t;
    }
  } else if (e < SZ + FEAT) {
    const int f = e - SZ;
    float run = 0.0f;
    for (int c = 0; c < NCHUNK; ++c) {
      const size_t ix = (size_t)(h * NCHUNK + c) * FEAT + f;
      const float t = zsum[ix];
      zsum[ix] = run;
      run += t;
    }
  }
}

// Intra-chunk attention: scores = Qp Kp^T (masked), attn = scores@V + Qp@Spre,
// denom = rowsum(masked scores) + qp . zpre.  One block per (chunk, head);
// 8 waves, each a 16x32 tile (2 accumulators per A fragment).
__global__ __launch_bounds__(256) void intra_attn_wmma(
    const float* __restrict__ qp, const float* __restrict__ kp,
    const float* __restrict__ qkv, const float* __restrict__ Spre,
    const float* __restrict__ zpre, float* __restrict__ attn) {
  __shared__ float sc[CHUNK * CHUNK];
  __shared__ float denom[CHUNK];
  const int c = blockIdx.x, h = blockIdx.y;
  const int c0 = c * CHUNK;
  const int tid = threadIdx.x;
  const int wave = tid >> 5, lane = tid & 31;
  const int half = lane >> 4, l = lane & 15;
  const int tm = (wave >> 1) << 4;   // row tile (0..48)
  const int tn0 = (wave & 1) << 5;   // col tile (0 or 32)
  const float* qpB = qp + ((size_t)h * NTOK + c0) * FEAT;
  const float* kpB = kp + ((size_t)h * NTOK + c0) * FEAT;

  // Phase 1: scores[i][j] = sum_f qp[i][f] kp[j][f]; causal mask j<=i.
  {
    const float* ap = qpB + (size_t)(tm + l) * FEAT + 2 * half;
    const float* bp0 = kpB + (size_t)(tn0 + l) * FEAT + 2 * half;  // B[f][j]
    const float* bp1 = bp0 + 16 * FEAT;
    v8f acc0 = {}, acc1 = {};
    for (int k = 0; k < FEAT; k += 4) {
      v2f a, b0, b1;
      a.x = ap[k];
      a.y = ap[k + 1];
      b0.x = bp0[k];
      b0.y = bp0[k + 1];
      b1.x = bp1[k];
      b1.y = bp1[k + 1];
      acc0 = wmma4(a, b0, acc0);
      acc1 = wmma4(a, b1, acc1);
    }
    const int row0 = tm + 8 * half;
#pragma unroll
    for (int i = 0; i < 8; ++i) {
      const int r = row0 + i;
      const int j0 = tn0 + l, j1 = tn0 + 16 + l;
      sc[r * CHUNK + j0] = (r >= j0) ? acc0[i] : 0.0f;
      sc[r * CHUNK + j1] = (r >= j1) ? acc1[i] : 0.0f;
    }
  }
  __syncthreads();

  // Denominator per row (waves 0-1 fully active -> no intra-wave divergence).
  if (tid < CHUNK) {
    float s = 0.0f;
    for (int j = 0; j < CHUNK; ++j) s += sc[tid * CHUNK + j];
    const float* q = qpB + (size_t)tid * FEAT;
    const float* z = zpre + (size_t)(h * NCHUNK + c) * FEAT;
    for (int f = 0; f < FEAT; ++f) s += q[f] * z[f];
    denom[tid] = s + EPSV;
  }
  __syncthreads();

  // Phase 2: attn_c = masked_scores @ V + Qp @ Spre; divide by denom.
  const float* vB = qkv + (size_t)c0 * QKV_LD + 2 * DIMX + h * HDIM;
  const float* SB = Spre + (size_t)(h * NCHUNK + c) * FEAT * HDIM;
  v8f acc0 = {}, acc1 = {};
  {  // intra: sc(64x64) @ V(64x64)
    const float* aL = sc + (tm + l) * CHUNK + 2 * half;
    const float* bp = vB + (size_t)(2 * half) * QKV_LD + tn0 + l;
    for (int k = 0; k < CHUNK; k += 4) {
      v2f a, b0, b1;
      a.x = aL[k];
      a.y = aL[k + 1];
      b0.x = bp[0];
      b0.y = bp[QKV_LD];
      b1.x = bp[16];
      b1.y = bp[QKV_LD + 16];
      acc0 = wmma4(a, b0, acc0);
      acc1 = wmma4(a, b1, acc1);
      bp += 4 * QKV_LD;
    }
  }
  {  // cross: qp(64x128) @ Spre(128x64)
    const float* ap = qpB + (size_t)(tm + l) * FEAT + 2 * half;
    const float* bp = SB + (size_t)(2 * half) * HDIM + tn0 + l;
    for (int k = 0; k < FEAT; k += 4) {
      v2f a, b0, b1;
      a.x = ap[k];
      a.y = ap[k + 1];
      b0.x = bp[0];
      b0.y = bp[HDIM];
      b1.x = bp[16];
      b1.y = bp[HDIM + 16];
      acc0 = wmma4(a, b0, acc0);
      acc1 = wmma4(a, b1, acc1);
      bp += 4 * HDIM;
    }
  }
  const int row0 = tm + 8 * half;
#pragma unroll
  for (int i = 0; i < 8; ++i) {
    const int r = row0 + i;
    const float inv = 1.0f / denom[r];
    attn[(size_t)(c0 + r) * DIMX + h * HDIM + tn0 + l] = acc0[i] * inv;
    attn[(size_t)(c0 + r) * DIMX + h * HDIM + tn0 + 16 + l] = acc1[i] * inv;
  }
}

extern "C" void kernel_launch(void* const* d_in, const int* in_sizes, int n_in,
                              void* d_out, int out_size, void* d_ws,
                              size_t ws_size, hipStream_t stream) {
  const float* x = (const float*)d_in[0];
  const float* proj = (const float*)d_in[1];
  const float* Wqkv = (const float*)d_in[2];
  const float* Wout = (const float*)d_in[3];
  const float* bout = (const float*)d_in[4];
  float* out = (float*)d_out;

  // Workspace layout (floats): ~42 MB total.
  float* ws = (float*)d_ws;
  float* qkv = ws;                                            // 2048*1536
  float* qp = qkv + (size_t)NTOK * QKV_LD;                    // 8*2048*128
  float* kp = qp + (size_t)HEADS * NTOK * FEAT;               // 8*2048*128
  float* Ssum = kp + (size_t)HEADS * NTOK * FEAT;             // 8*32*128*64
  float* zsum = Ssum + (size_t)HEADS * NCHUNK * FEAT * HDIM;  // 8*32*128
  float* attn = zsum + (size_t)HEADS * NCHUNK * FEAT;         // 2048*512

  // 1) qkv = x @ W_qkv   (2048 x 1536, K=512): 128*24 = 3072 waves
  gemm_wmma_f32<NTOK, QKV_LD, DIMX, DIMX, QKV_LD, QKV_LD, false>
      <<<3072 / 8, 256, 0, stream>>>(x, Wqkv, qkv, nullptr);

  // 2) qp/kp = elu(head @ proj)+1  (16 batched 2048x128, K=64): 256 waves each
  {
    dim3 g(256 / 8, 16);
    featmap_wmma<<<g, 256, 0, stream>>>(qkv, proj, qp, kp);
  }
  // 3) per-chunk Kp^T @ V and sum(Kp)
  {
    dim3 g(NCHUNK, HEADS);
    chunk_kv_wmma<<<g, 256, 0, stream>>>(kp, qkv, Ssum);
    chunk_z<<<(HEADS * NCHUNK * FEAT + 255) / 256, 256, 0, stream>>>(kp, zsum);
  }
  // 4) exclusive prefix over chunks
  {
    dim3 g((FEAT * HDIM + FEAT + 255) / 256, HEADS);
    prefix_chunks<<<g, 256, 0, stream>>>(Ssum, zsum);
  }
  // 5) intra-chunk attention + normalization
  {
    dim3 g(NCHUNK, HEADS);
    intra_attn_wmma<<<g, 256, 0, stream>>>(qp, kp, qkv, Ssum, zsum, attn);
  }
  // 6) out = attn @ W_out + b_out  (2048 x 512, K=512): 128*8 = 1024 waves
  gemm_wmma_f32<NTOK, DIMX, DIMX, DIMX, DIMX, DIMX, true>
      <<<1024 / 8, 256, 0, stream>>>(attn, Wout, out, bout);
}